// MemoryGraph_42210938585856
// MI455X (gfx1250) — compile-verified
//
#include <hip/hip_runtime.h>
#include <math.h>

// ---------------------------------------------------------------------------
// Problem dimensions (fixed by the reference)
// ---------------------------------------------------------------------------
constexpr int kBS      = 64;    // batch
constexpr int kNC      = 128;   // cells
constexpr int kN       = 32;    // neurons / cell
constexpr int kDN      = 64;    // per-neuron dim
constexpr int kHMOD    = 128;   // hidden
constexpr int kMODIN   = 2 * kN + 4 + kN * kN;   // 1092
constexpr int kMODOUT  = kN * kN + kN;           // 1056
constexpr int kHEAD    = 2 * kN + 4;             // 68 non-hebbian inputs
constexpr int kHSTRIDE = 132;   // padded LDS stride for hidden (132 % 64 = 4)

typedef __attribute__((ext_vector_type(2))) float v2f;
typedef __attribute__((ext_vector_type(8))) float v8f;

// ---------------------------------------------------------------------------
// Kernel 1: build the 68-element "head" of mod_input per (b, cell):
//   [0..31]  ||h[b,c,i,:]||      [32..63] ||msg[b,c,i,:]||
//   [64] mean(decay_logit)  [65] readout_drift  [66] s_live  [67] s_ema
// One thread per norm; thread j==0 of each pair also writes the tail.
// ---------------------------------------------------------------------------
__global__ __launch_bounds__(256) void head_kernel(
    const float* __restrict__ h, const float* __restrict__ msg,
    const float* __restrict__ decay, const float* __restrict__ drift,
    const float* __restrict__ s_live, const float* __restrict__ s_ema,
    float* __restrict__ head) {
  const int t    = blockIdx.x * 256 + threadIdx.x;   // < BS*NC*64
  const int pair = t >> 6;                           // (b * NC + c)
  const int j    = t & 63;

  const float* src = (j < kN ? h : msg) +
                     (size_t)pair * (kN * kDN) + (size_t)(j & (kN - 1)) * kDN;
  const float4* p4 = (const float4*)src;
  float s = 0.f;
#pragma unroll
  for (int i = 0; i < kDN / 4; ++i) {
    float4 v = p4[i];
    s += v.x * v.x + v.y * v.y + v.z * v.z + v.w * v.w;
  }
  head[(size_t)pair * kHEAD + j] = sqrtf(s);

  if (j == 0) {
    const float* dp = decay + (size_t)pair * kN;
    float m = 0.f;
#pragma unroll
    for (int i = 0; i < kN; ++i) m += dp[i];
    const int b = pair / kNC;
    float* hd = head + (size_t)pair * kHEAD;
    hd[kHEAD - 4] = m * (1.f / (float)kN);
    hd[kHEAD - 3] = drift[pair];
    hd[kHEAD - 2] = s_live[b];
    hd[kHEAD - 1] = s_ema[b];
  }
}

// ---------------------------------------------------------------------------
// Kernel 2: per-cell fused 2-layer MLP using V_WMMA_F32_16X16X4_F32.
//   grid = (BS/16 m-tiles, NC cells); block = 256 threads = 8 wave32.
//   LDS: A tile 16x1092 (mod_input rows) + hidden 16x132.
// Fragment layouts (ISA 7.12.2, 32-bit 16x4 A / 16x16 C):
//   A: lane l, row M = l&15; VGPR0/1 hold K = k0 + 2*(l>>4) + {0,1}
//   B: lane l, col N = l&15; VGPR0/1 hold K = k0 + 2*(l>>4) + {0,1}
//   C/D: VGPR i -> row M = i + 8*(l>>4), col N = l&15
// ---------------------------------------------------------------------------
__global__ __launch_bounds__(256) void modmlp_kernel(
    const float* __restrict__ head, const float* __restrict__ hebb,
    const float* __restrict__ w1, const float* __restrict__ b1,
    const float* __restrict__ w2, const float* __restrict__ b2,
    float* __restrict__ out) {
  extern __shared__ float smem[];
  float* ldsA = smem;                    // 16 * 1092 floats
  float* ldsH = smem + 16 * kMODIN;      // 16 * 132  floats

  const int tid   = threadIdx.x;
  const int lane  = tid & 31;
  const int wave  = tid >> 5;
  const int mtile = blockIdx.x;          // batch tile (rows b0..b0+15)
  const int cell  = blockIdx.y;
  const int b0    = mtile * 16;
  const int arow  = lane & 15;           // M within tile
  const int koff  = (lane >> 4) << 1;    // 0 or 2 (K sub-offset)
  const int rhalf = (lane >> 4) << 3;    // 0 or 8 (C-row offset)

  // ---- Stage A = mod_input[b0..b0+15][0..1091] into LDS -------------------
  for (int r = 0; r < 16; ++r) {
    const size_t pr = (size_t)(b0 + r) * kNC + cell;
    const float* hr = head + pr * kHEAD;
    const float* br = hebb + pr * (kN * kN);
    float* dst = ldsA + r * kMODIN;
    if (tid < kHEAD) dst[tid] = hr[tid];
    for (int k = tid; k < kN * kN; k += 256) dst[kHEAD + k] = br[k];
  }
  __syncthreads();

  // ---- Phase 1: hidden(16x128) = tanh(A(16x1092) * w1 + b1) ---------------
  // wave w owns hidden columns [16w, 16w+16)
  const float* w1c  = w1 + (size_t)cell * kMODIN * kHMOD;
  const int    hcol = (wave << 4) + arow;
  v8f acc;
  {
    const float bias = b1[cell * kHMOD + hcol];   // col is per-lane constant
#pragma unroll
    for (int i = 0; i < 8; ++i) acc[i] = bias;
  }
  const float* aRow = ldsA + arow * kMODIN + koff;
#pragma unroll 4
  for (int k0 = 0; k0 < kMODIN; k0 += 4) {
    const v2f a = *(const v2f*)(aRow + k0);       // ds_load_b64
    v2f b;
    b.x = w1c[(size_t)(k0 + koff) * kHMOD + hcol];
    b.y = w1c[(size_t)(k0 + koff + 1) * kHMOD + hcol];
    acc = __builtin_amdgcn_wmma_f32_16x16x4_f32(
        false, a, false, b, (short)0, acc, false, false);
  }
#pragma unroll
  for (int i = 0; i < 8; ++i)
    ldsH[(i + rhalf) * kHSTRIDE + hcol] = tanhf(acc[i]);
  __syncthreads();

  // ---- Phase 2: out(16x1056) = hidden(16x128) * w2 + b2 -------------------
  const float* w2c = w2 + (size_t)cell * kHMOD * kMODOUT;
  const float* b2c = b2 + (size_t)cell * kMODOUT;
  const float* aH  = ldsH + arow * kHSTRIDE + koff;
  for (int ot = wave; ot < kMODOUT / 16; ot += 8) {   // wave-uniform loop
    const int ocol = (ot << 4) + arow;
    v8f c;
    {
      const float bias = b2c[ocol];
#pragma unroll
      for (int i = 0; i < 8; ++i) c[i] = bias;
    }
#pragma unroll 8
    for (int k0 = 0; k0 < kHMOD; k0 += 4) {
      const v2f a = *(const v2f*)(aH + k0);       // ds_load_b64
      v2f b;
      b.x = w2c[(size_t)(k0 + koff) * kMODOUT + ocol];
      b.y = w2c[(size_t)(k0 + koff + 1) * kMODOUT + ocol];
      c = __builtin_amdgcn_wmma_f32_16x16x4_f32(
          false, a, false, b, (short)0, c, false, false);
    }
#pragma unroll
    for (int i = 0; i < 8; ++i) {
      const int row = b0 + rhalf + i;
      out[((size_t)row * kNC + cell) * kMODOUT + ocol] = c[i];
    }
  }
}

// ---------------------------------------------------------------------------
extern "C" void kernel_launch(void* const* d_in, const int* in_sizes, int n_in,
                              void* d_out, int out_size, void* d_ws, size_t ws_size,
                              hipStream_t stream) {
  const float* h     = (const float*)d_in[0];
  const float* msg   = (const float*)d_in[1];
  const float* decay = (const float*)d_in[2];
  const float* hebb  = (const float*)d_in[3];
  const float* drift = (const float*)d_in[4];
  const float* slive = (const float*)d_in[5];
  const float* sema  = (const float*)d_in[6];
  const float* w1    = (const float*)d_in[7];
  const float* b1    = (const float*)d_in[8];
  const float* w2    = (const float*)d_in[9];
  const float* b2    = (const float*)d_in[10];
  float* headws = (float*)d_ws;              // BS*NC*68 floats (~2.2 MB)
  float* out    = (float*)d_out;

  const int norm_threads = kBS * kNC * 2 * kN;           // 524288
  head_kernel<<<norm_threads / 256, 256, 0, stream>>>(
      h, msg, decay, drift, slive, sema, headws);

  dim3 grid(kBS / 16, kNC);                              // 4 x 128
  const size_t smem = (size_t)(16 * kMODIN + 16 * kHSTRIDE) * sizeof(float);
  modmlp_kernel<<<grid, 256, smem, stream>>>(
      headws, hebb, w1, b1, w2, b2, out);
}